// MinCountLoss_15539191677200
// MI455X (gfx1250) — compile-verified
//
#include <hip/hip_runtime.h>

typedef float v2f __attribute__((ext_vector_type(2)));
typedef float v8f __attribute__((ext_vector_type(8)));

#define BDIM 32
#define HDIM 1024
#define WDIM 1024
#define NBOX 96
#define CH 8                  // column chunks per image
#define CHROWS (HDIM / CH)    // 128 rows per chunk

// Uniform constant-lane broadcast: v_readlane_b32 (SGPR result), no LDS path.
__device__ __forceinline__ float readlane_f32(float v, int lane) {
  return __uint_as_float(__builtin_amdgcn_readlane(__float_as_uint(v), lane));
}

// ---------------------------------------------------------------------------
// Kernel 1: row-wise inclusive prefix sum.
// One block per row (b*H + h), 4 waves per block, each wave scans a 256-float
// segment as a 16x16 tile using 8 chained V_WMMA_F32_16X16X4_F32:
//   Y = X * U          (U = upper-triangular ones  -> within-row-of-16 scans)
//   F = Ls * T + Y     (Ls = strict lower ones, T = row totals -> offsets)
// ---------------------------------------------------------------------------
__global__ __launch_bounds__(128) void row_scan_wmma(const float* __restrict__ in,
                                                     float* __restrict__ out) {
  __shared__ float wtot[4];
  const int row  = blockIdx.x;            // b*H + h
  const int lane = threadIdx.x & 31;
  const int wave = threadIdx.x >> 5;
  const float* seg = in  + (size_t)row * WDIM + wave * 256;
  float*       dst = out + (size_t)row * WDIM + wave * 256;

  const int m  = lane & 15;               // A: tile row   | B/C/D: tile column
  const int h2 = (lane >> 4) << 1;        // 0 (lanes 0-15) / 2 (lanes 16-31)

  // ---- Y = X * U : within-16 inclusive scan of each of the 16 rows ----
  v8f y = {0.f, 0.f, 0.f, 0.f, 0.f, 0.f, 0.f, 0.f};
#pragma unroll
  for (int k = 0; k < 4; ++k) {
    const int c0 = 4 * k + h2;
    v2f a;                                  // A = X[:, c0:c0+2] per half-wave
    a.x = seg[m * 16 + c0];
    a.y = seg[m * 16 + c0 + 1];
    v2f b;                                  // B = U[c0:c0+2, :], col = m
    b.x = (c0     <= m) ? 1.f : 0.f;
    b.y = (c0 + 1 <= m) ? 1.f : 0.f;
    y = __builtin_amdgcn_wmma_f32_16x16x4_f32(false, a, false, b,
                                              (short)0, y, false, false);
  }

  // ---- gather 16 row totals t[r] = Y[r][15]: constant-lane readlane (SGPRs)
  float t[16];
#pragma unroll
  for (int r = 0; r < 8; ++r) {
    t[r]     = readlane_f32(y[r], 15);      // rows 0..7  live in lanes 0-15
    t[r + 8] = readlane_f32(y[r], 31);      // rows 8..15 live in lanes 16-31
  }

  // ---- F = Ls * T + Y : add exclusive running offsets across rows ----
  const bool hi = (lane >= 16);
  v8f f = y;
#pragma unroll
  for (int k = 0; k < 4; ++k) {
    const int c0 = 4 * k + h2;
    v2f a;                                  // A = Ls[m, c0:c0+2]
    a.x = (c0     < m) ? 1.f : 0.f;
    a.y = (c0 + 1 < m) ? 1.f : 0.f;
    v2f b;                                  // B = T[c0:c0+2, :] (row-broadcast)
    b.x = hi ? t[4 * k + 2] : t[4 * k];     // one cndmask of two SGPRs
    b.y = hi ? t[4 * k + 3] : t[4 * k + 1];
    f = __builtin_amdgcn_wmma_f32_16x16x4_f32(false, a, false, b,
                                              (short)0, f, false, false);
  }

  // ---- inter-wave carry within the 1024-wide row ----
  const float wsum = readlane_f32(f[7], 31);  // F[15][15] = segment total
  if (lane == 0) wtot[wave] = wsum;
  __syncthreads();
  float carry = 0.f;
#pragma unroll
  for (int w = 0; w < 4; ++w) carry += (w < wave) ? wtot[w] : 0.f;

#pragma unroll
  for (int i = 0; i < 8; ++i) {
    const int r = i + ((lane >> 4) << 3);   // rows 0-7 / 8-15 per half-wave
    dst[r * 16 + m] = f[i] + carry;
  }
}

// ---------------------------------------------------------------------------
// Kernel 2a: in-place column scan inside each 128-row chunk; chunk totals out.
// tid = (b*CH + chunk)*W + x  -> lanes map to consecutive x: coalesced.
// ---------------------------------------------------------------------------
__global__ __launch_bounds__(256) void col_chunk_scan(float* __restrict__ s,
                                                      float* __restrict__ tot) {
  const int tid   = blockIdx.x * blockDim.x + threadIdx.x;
  const int x     = tid % WDIM;
  const int bc    = tid / WDIM;
  const int chunk = bc % CH;
  const int b     = bc / CH;
  float* p = s + (size_t)b * HDIM * WDIM + (size_t)chunk * CHROWS * WDIM + x;
  float acc = 0.f;
  for (int r = 0; r < CHROWS; ++r) {
    acc += p[(size_t)r * WDIM];
    p[(size_t)r * WDIM] = acc;
  }
  tot[tid] = acc;
}

// ---------------------------------------------------------------------------
// Kernel 2b: add exclusive prefix of chunk totals into chunks 1..CH-1.
// ---------------------------------------------------------------------------
__global__ __launch_bounds__(256) void col_chunk_fixup(float* __restrict__ s,
                                                       const float* __restrict__ tot) {
  const int tid   = blockIdx.x * blockDim.x + threadIdx.x;
  const int x     = tid % WDIM;
  const int bc    = tid / WDIM;
  const int chunk = bc % CH;
  const int b     = bc / CH;
  if (chunk == 0) return;
  float carry = 0.f;
  for (int c = 0; c < chunk; ++c) carry += tot[(size_t)(b * CH + c) * WDIM + x];
  float* p = s + (size_t)b * HDIM * WDIM + (size_t)chunk * CHROWS * WDIM + x;
  for (int r = 0; r < CHROWS; ++r) p[(size_t)r * WDIM] += carry;
}

// ---------------------------------------------------------------------------
// Kernel 3: gather SAT corners for all B*N boxes, relu(1 - box_sum), reduce.
// S (in ws) holds inclusive 2-D cumsum: padded S[y][x] == ws[y-1][x-1],
// with S(0,.) = S(.,0) = 0. clip is to [0, W] inclusive (matches jnp.clip).
// ---------------------------------------------------------------------------
__global__ __launch_bounds__(1024) void box_loss(const float* __restrict__ s,
                                                 const int* __restrict__ bb,
                                                 float* __restrict__ out) {
  __shared__ float red[1024];
  float acc = 0.f;
  for (int i = threadIdx.x; i < BDIM * NBOX; i += 1024) {
    const int b = i / NBOX;
    int x1 = bb[i * 4 + 0], y1 = bb[i * 4 + 1];
    int x2 = bb[i * 4 + 2], y2 = bb[i * 4 + 3];
    x1 = min(max(x1, 0), WDIM); y1 = min(max(y1, 0), WDIM);
    x2 = min(max(x2, 0), WDIM); y2 = min(max(y2, 0), WDIM);
    const float* S = s + (size_t)b * HDIM * WDIM;
    auto g = [&](int yy, int xx) -> float {
      return (yy > 0 && xx > 0) ? S[(size_t)(yy - 1) * WDIM + (xx - 1)] : 0.f;
    };
    float bs = g(y2, x2) - g(y1, x2) - g(y2, x1) + g(y1, x1);
    if (!((x2 > x1) && (y2 > y1))) bs = 0.f;
    const float l = 1.f - bs;
    acc += (l > 0.f) ? l : 0.f;
  }
  red[threadIdx.x] = acc;
  __syncthreads();
  for (int stride = 512; stride > 0; stride >>= 1) {
    if ((int)threadIdx.x < stride) red[threadIdx.x] += red[threadIdx.x + stride];
    __syncthreads();
  }
  if (threadIdx.x == 0) out[0] = red[0];
}

// ---------------------------------------------------------------------------
// Launch: K1 row scan (WMMA) -> K2a/K2b column scan -> K3 box loss.
// Workspace layout: [ SAT : B*H*W f32 = 128 MB ][ chunk totals : B*CH*W f32 ].
// ---------------------------------------------------------------------------
extern "C" void kernel_launch(void* const* d_in, const int* in_sizes, int n_in,
                              void* d_out, int out_size, void* d_ws, size_t ws_size,
                              hipStream_t stream) {
  (void)in_sizes; (void)n_in; (void)out_size; (void)ws_size;
  const float* img = (const float*)d_in[0];
  const int*   bb  = (const int*)d_in[1];
  float* S   = (float*)d_ws;
  float* tot = S + (size_t)BDIM * HDIM * WDIM;

  row_scan_wmma<<<BDIM * HDIM, 128, 0, stream>>>(img, S);

  const int n2 = BDIM * CH * WDIM;            // 262144 threads
  col_chunk_scan <<<n2 / 256, 256, 0, stream>>>(S, tot);
  col_chunk_fixup<<<n2 / 256, 256, 0, stream>>>(S, tot);

  box_loss<<<1, 1024, 0, stream>>>(S, bb, (float*)d_out);
}